// PEGH_43026982371475
// MI455X (gfx1250) — compile-verified
//
#include <hip/hip_runtime.h>
#include <stdint.h>

// ---------------------------------------------------------------------------
// Sparse depthwise 3x3 conv over a 256x256 grid, 512 channels, fp32.
// Bandwidth-bound (~85 MB HBM-side; x resident in the 192MB L2). No matmul
// structure (weight AND activation both indexed by channel -> no shared
// contraction dim), so WMMA is inapplicable; instead we use the CDNA5
// data-movement paths: async global->LDS staging (ASYNCcnt), s_wait_asynccnt,
// global_prefetch_b8, wave32-safe code, b128 vector memory ops.
// ---------------------------------------------------------------------------

#define DIMC 512
#define GH 256
#define GW 256
#define NCELLS (GH * GW)
#define CPB 16            // cells per block in the conv kernel
#define CONV_THREADS 128  // 128 threads * float4 = 512 channels

#if __has_builtin(__builtin_amdgcn_global_load_async_to_lds_b128)
#define HAVE_ASYNC_LDS 1
#else
#define HAVE_ASYNC_LDS 0
#endif

#if HAVE_ASYNC_LDS
// Builtin signature (confirmed by hipcc diagnostics): (v4i as1*, v4i as3*, Imm, Imm)
typedef int v4i __attribute__((vector_size(4 * sizeof(int))));
typedef __attribute__((address_space(1))) v4i* g_v4i_ptr;
typedef __attribute__((address_space(3))) v4i* l_v4i_ptr;
#endif

// --- init: idx_grid = -1, min2 = INT_MAX ----------------------------------
__global__ void PEGH_init_kernel(int* __restrict__ idx_grid, int* __restrict__ min2) {
  int k = blockIdx.x * 256 + threadIdx.x;
  if (k < NCELLS) idx_grid[k] = -1;
  if (k < 2) min2[k] = 0x7fffffff;
}

// --- weight transpose: w[c*9 + t] -> wT[t*512 + c] ------------------------
__global__ void PEGH_wtrans_kernel(const float* __restrict__ w, float* __restrict__ wT) {
  int k = blockIdx.x * 256 + threadIdx.x;
  if (k < 9 * DIMC) {
    int c = k / 9, t = k - c * 9;
    wT[t * DIMC + c] = w[k];
  }
}

// --- position minimum (reference subtracts pos.min(axis=0)) ---------------
__global__ void PEGH_posmin_kernel(const int* __restrict__ pos, int* __restrict__ min2, int nPts) {
  int i = blockIdx.x * 256 + threadIdx.x;
  if (i < nPts) {
    atomicMin(&min2[0], pos[2 * i + 0]);
    atomicMin(&min2[1], pos[2 * i + 1]);
  }
}

// --- scatter point index into grid ----------------------------------------
__global__ void PEGH_scatter_kernel(const int* __restrict__ pos, const int* __restrict__ min2,
                                    int* __restrict__ idx_grid, int nPts) {
  int i = blockIdx.x * 256 + threadIdx.x;
  if (i < nPts) {
    int p0 = pos[2 * i + 0] - min2[0];
    int p1 = pos[2 * i + 1] - min2[1];
    idx_grid[(p0 << 8) | p1] = i;  // positions are unique
  }
}

// --- block-wide inclusive scan helper (256 threads) ------------------------
__device__ __forceinline__ int PEGH_block_scan256(int v, int* s) {
  const int tid = threadIdx.x;
  s[tid] = v;
  __syncthreads();
#pragma unroll
  for (int off = 1; off < 256; off <<= 1) {
    int cur = s[tid];
    int add = (tid >= off) ? s[tid - off] : 0;
    __syncthreads();
    s[tid] = cur + add;
    __syncthreads();
  }
  return s[tid];  // inclusive prefix of v
}

// --- phase 1: per-block occupancy counts (256 cells/block, coalesced) ------
__global__ __launch_bounds__(256) void PEGH_count_kernel(const int* __restrict__ idx_grid,
                                                         int* __restrict__ blk_cnt) {
  __shared__ int s[256];
  int idx = blockIdx.x * 256 + threadIdx.x;
  int occ = (idx_grid[idx] >= 0) ? 1 : 0;
  int incl = PEGH_block_scan256(occ, s);
  if (threadIdx.x == 255) blk_cnt[blockIdx.x] = incl;  // block total
}

// --- phase 2: exclusive scan of the 256 block totals (one block) -----------
__global__ __launch_bounds__(256) void PEGH_scanblk_kernel(const int* __restrict__ blk_cnt,
                                                           int* __restrict__ blk_base) {
  __shared__ int s[256];
  int v = blk_cnt[threadIdx.x];
  int incl = PEGH_block_scan256(v, s);
  blk_base[threadIdx.x] = incl - v;  // exclusive base
}

// --- phase 3: emit cell_of_rank in flat (lexsort) order --------------------
__global__ __launch_bounds__(256) void PEGH_emit_kernel(const int* __restrict__ idx_grid,
                                                        const int* __restrict__ blk_base,
                                                        int* __restrict__ cell_of_rank) {
  __shared__ int s[256];
  int idx = blockIdx.x * 256 + threadIdx.x;
  int occ = (idx_grid[idx] >= 0) ? 1 : 0;
  int incl = PEGH_block_scan256(occ, s);
  if (occ) cell_of_rank[blk_base[blockIdx.x] + incl - 1] = idx;
}

// --- main conv/gather kernel ----------------------------------------------
__global__ __launch_bounds__(CONV_THREADS) void PEGH_conv_kernel(
    const float* __restrict__ x, const float* __restrict__ wT, const float* __restrict__ bias,
    const int* __restrict__ idx_grid, const int* __restrict__ cell_of_rank,
    float* __restrict__ out, int nPts) {
  __shared__ __align__(16) float s_w[9 * DIMC];   // 18 KB, [tap][channel]
  __shared__ __align__(16) float s_bias[DIMC];    // 2 KB
  __shared__ int s_nbr[CPB][9];                   // neighbor point indices

  const int tid = threadIdx.x;

  // Stage weights + bias to LDS via CDNA5 async global->LDS DMA
#if HAVE_ASYNC_LDS
  for (int k = tid; k < (9 * DIMC) / 4; k += CONV_THREADS)
    __builtin_amdgcn_global_load_async_to_lds_b128(
        (g_v4i_ptr)(wT + 4 * k), (l_v4i_ptr)(s_w + 4 * k), 0, 0);
  for (int k = tid; k < DIMC / 4; k += CONV_THREADS)
    __builtin_amdgcn_global_load_async_to_lds_b128(
        (g_v4i_ptr)(bias + 4 * k), (l_v4i_ptr)(s_bias + 4 * k), 0, 0);
#else
  for (int k = tid; k < 9 * DIMC; k += CONV_THREADS) s_w[k] = wT[k];
  for (int k = tid; k < DIMC; k += CONV_THREADS) s_bias[k] = bias[k];
#endif

  // Resolve the (<=9) occupied neighbor point indices for each of our cells,
  // prefetching their feature rows while the weight DMA is in flight.
  const int r0 = blockIdx.x * CPB;
  for (int k = tid; k < CPB * 9; k += CONV_THREADS) {
    int s = k / 9, t = k - s * 9;
    int nb = -1;
    int r = r0 + s;
    if (r < nPts) {
      int cell = cell_of_rank[r];
      int p0 = cell >> 8, p1 = cell & 255;
      int q0 = p0 + (t % 3) - 1;  // kernel index j walks p0 (transposed conv)
      int q1 = p1 + (t / 3) - 1;  // kernel index i walks p1
      if (q0 >= 0 && q0 < GH && q1 >= 0 && q1 < GW) nb = idx_grid[(q0 << 8) | q1];
      if (nb >= 0) __builtin_prefetch(x + (size_t)nb * DIMC, 0, 1);
    }
    s_nbr[s][t] = nb;
  }

#if HAVE_ASYNC_LDS
#if __has_builtin(__builtin_amdgcn_s_wait_asynccnt)
  __builtin_amdgcn_s_wait_asynccnt(0);
#else
  asm volatile("s_wait_asynccnt 0" ::: "memory");
#endif
#endif
  __syncthreads();

  const int c0 = tid * 4;  // this thread's 4 channels
  const float4 b4 = *(const float4*)(s_bias + c0);
  const int smax = min(CPB, nPts - r0);

  for (int s = 0; s < smax; ++s) {
    const int r = r0 + s;
    // s_nbr values are block-uniform: lift them to SGPRs for scalar branches
    const int ic = __builtin_amdgcn_readfirstlane(s_nbr[s][4]);  // center
    float4 acc = b4;
    const float4 xc = *(const float4*)(x + (size_t)ic * DIMC + c0);
    acc.x += xc.x; acc.y += xc.y; acc.z += xc.z; acc.w += xc.w;  // identity
#pragma unroll
    for (int t = 0; t < 9; ++t) {
      const int jn = __builtin_amdgcn_readfirstlane(s_nbr[s][t]);
      if (jn >= 0) {
        const float4 w4 = *(const float4*)(s_w + t * DIMC + c0);
        const float4 xv = *(const float4*)(x + (size_t)jn * DIMC + c0);
        acc.x = fmaf(w4.x, xv.x, acc.x);
        acc.y = fmaf(w4.y, xv.y, acc.y);
        acc.z = fmaf(w4.z, xv.z, acc.z);
        acc.w = fmaf(w4.w, xv.w, acc.w);
      }
    }
    *(float4*)(out + (size_t)r * DIMC + c0) = acc;  // coalesced b128 store
  }
}

extern "C" void kernel_launch(void* const* d_in, const int* in_sizes, int n_in,
                              void* d_out, int out_size, void* d_ws, size_t ws_size,
                              hipStream_t stream) {
  const float* x = (const float*)d_in[0];     // (1, N, 512)
  const int* pos = (const int*)d_in[1];       // (N, 2)
  const float* w = (const float*)d_in[2];     // (512, 1, 3, 3)
  const float* bias = (const float*)d_in[3];  // (512,)
  float* out = (float*)d_out;                 // (1, N, 512)
  const int nPts = in_sizes[1] / 2;

  // Workspace layout
  char* ws = (char*)d_ws;
  int* idx_grid = (int*)ws;                                   // 65536 ints
  int* min2 = (int*)(ws + (size_t)NCELLS * 4);                // 2 ints
  int* blk_cnt = (int*)(ws + (size_t)NCELLS * 4 + 64);        // 256 ints
  int* blk_base = (int*)(ws + (size_t)NCELLS * 4 + 64 + 1024);// 256 ints
  float* wT = (float*)(ws + (size_t)NCELLS * 4 + 64 + 2048);  // 4608 floats
  int* cell_of_rank =
      (int*)(ws + (size_t)NCELLS * 4 + 64 + 2048 + (size_t)9 * DIMC * 4);  // nPts ints

  PEGH_init_kernel<<<(NCELLS + 255) / 256, 256, 0, stream>>>(idx_grid, min2);
  PEGH_wtrans_kernel<<<(9 * DIMC + 255) / 256, 256, 0, stream>>>(w, wT);
  PEGH_posmin_kernel<<<(nPts + 255) / 256, 256, 0, stream>>>(pos, min2, nPts);
  PEGH_scatter_kernel<<<(nPts + 255) / 256, 256, 0, stream>>>(pos, min2, idx_grid, nPts);
  PEGH_count_kernel<<<NCELLS / 256, 256, 0, stream>>>(idx_grid, blk_cnt);
  PEGH_scanblk_kernel<<<1, 256, 0, stream>>>(blk_cnt, blk_base);
  PEGH_emit_kernel<<<NCELLS / 256, 256, 0, stream>>>(idx_grid, blk_base, cell_of_rank);
  PEGH_conv_kernel<<<(nPts + CPB - 1) / CPB, CONV_THREADS, 0, stream>>>(
      x, wT, bias, idx_grid, cell_of_rank, out, nPts);
}